// Voxel_Wise_Mapping_87780541596086
// MI455X (gfx1250) — compile-verified
//
#include <hip/hip_runtime.h>

typedef __attribute__((ext_vector_type(2))) float v2f;
typedef __attribute__((ext_vector_type(4))) float f4;
typedef __attribute__((ext_vector_type(8))) float v8f;

#define IN_C  64
#define NSPLIT 8
#define ROWS_PER_WAVE 16
#define WAVES_PER_BLOCK 8

__global__ __launch_bounds__(256) void voxel_wise_mapping_kernel(
    const float* __restrict__ feat,   // [N, 64]
    const float* __restrict__ W,      // [64, 8]
    const float* __restrict__ bias,   // [8]
    float* __restrict__ out,          // [8, N, 64]
    int N)
{
    const int lane = threadIdx.x & 31;
    const int wave = threadIdx.x >> 5;
    const int tile = blockIdx.x * WAVES_PER_BLOCK + wave;
    const int r0   = tile * ROWS_PER_WAVE;
    if (r0 >= N) return;                 // wave-uniform

    const int n16 = lane & 15;           // column within 16-wide tile
    const int hi  = lane >> 4;           // half-wave select

    // ---------------- B fragments: W [64x8] zero-padded to [64x16] ----------
    // B 4x16 f32 layout: lanes 0-15 hold K=k0,k0+1 (v0,v1); lanes 16-31 K=k0+2,k0+3
    v2f Bf[16];
#pragma unroll
    for (int kk = 0; kk < 16; ++kk) {
        const int k0 = kk * 4 + hi * 2;
        float b0 = 0.0f, b1 = 0.0f;
        if (n16 < NSPLIT) {
            b0 = W[k0 * NSPLIT + n16];
            b1 = W[(k0 + 1) * NSPLIT + n16];
        }
        Bf[kk][0] = b0;
        Bf[kk][1] = b1;
    }

    // ---------------- C init with bias (padded cols get exactly 0) ----------
    const float bv = (n16 < NSPLIT) ? bias[n16] : 0.0f;
    v8f c;
#pragma unroll
    for (int i = 0; i < 8; ++i) c[i] = bv;

    // ---------------- A loads + 16 x WMMA f32 16x16x4 (K = 64) --------------
    int rowA = r0 + n16;
    if (rowA > N - 1) rowA = N - 1;      // clamp for tail tiles (keeps EXEC full)
    const float* frow = feat + (size_t)rowA * IN_C;
#pragma unroll
    for (int kk = 0; kk < 16; ++kk) {
        const int k0 = kk * 4 + hi * 2;
        v2f a;
        a[0] = frow[k0];
        a[1] = frow[k0 + 1];
        // 8 args: (neg_a, A, neg_b, B, c_mod, C, reuse_a, reuse_b)
        c = __builtin_amdgcn_wmma_f32_16x16x4_f32(
                false, a, false, Bf[kk], (short)0, c, false, false);
    }

    // ---------------- per-row argmax over n = 0..7 ---------------------------
    // Row v logits live in c[v]: lanes 0-7 (rows 0..7) / lanes 16-23 (rows 8..15).
    // Width-8 butterfly; tie-break = smallest index (first max, like jnp.argmax).
    int cls[8];
#pragma unroll
    for (int v = 0; v < 8; ++v) {
        float val = c[v];
        int   idx = lane & 7;
#pragma unroll
        for (int off = 4; off >= 1; off >>= 1) {
            float ov = __shfl_xor(val, off, 8);
            int   oi = __shfl_xor(idx, off, 8);
            if (ov > val || (ov == val && oi < idx)) { val = ov; idx = oi; }
        }
        cls[v] = idx;   // lane 0 -> class(row v), lane 16 -> class(row v+8)
    }

    // ---------------- scatter/store phase ------------------------------------
    // out[n, row, :] = (n == class(row)) ? feat[row, :] : 0
    // Each lane owns float4 column c4 = lane&15 of the row, and slabs
    // n = 2*it + hi for it = 0..3. All stores non-temporal b128 (pure stream).
    const f4* feat4 = (const f4*)feat;
    f4*       out4  = (f4*)out;
    const int c4    = lane & 15;
    const f4  zero4 = (f4){0.0f, 0.0f, 0.0f, 0.0f};

#pragma unroll
    for (int m = 0; m < ROWS_PER_WAVE; ++m) {
        const int row = r0 + m;
        if (row >= N) break;             // wave-uniform
        const int km = __shfl(cls[m & 7], (m < 8) ? 0 : 16, 32);
        const f4 fv = feat4[(size_t)row * 16 + c4];
#pragma unroll
        for (int it = 0; it < 4; ++it) {
            const int n = it * 2 + hi;
            const f4 val = (n == km) ? fv : zero4;
            __builtin_nontemporal_store(
                val, &out4[((size_t)n * (size_t)N + (size_t)row) * 16 + c4]);
        }
    }
}

extern "C" void kernel_launch(void* const* d_in, const int* in_sizes, int n_in,
                              void* d_out, int out_size, void* d_ws, size_t ws_size,
                              hipStream_t stream) {
    const float* feat = (const float*)d_in[0];   // [N, 64] f32
    const float* W    = (const float*)d_in[1];   // [64, 8] f32
    const float* b    = (const float*)d_in[2];   // [8]     f32
    float*       out  = (float*)d_out;           // [8, N, 64] f32

    const int N = in_sizes[0] / IN_C;
    const int tiles  = (N + ROWS_PER_WAVE - 1) / ROWS_PER_WAVE;
    const int blocks = (tiles + WAVES_PER_BLOCK - 1) / WAVES_PER_BLOCK;

    voxel_wise_mapping_kernel<<<blocks, 256, 0, stream>>>(feat, W, b, out, N);
}